// VQExpert_75076028334462
// MI455X (gfx1250) — compile-verified
//
#include <hip/hip_runtime.h>

#define IN_FEAT   2048
#define HIDDEN    1024
#define OUT_FEAT  2048
#define CB_DIM    32
#define NUM_CODES 256
#define TOKENS    16384   // B*S = 4*4096

typedef float v2f __attribute__((ext_vector_type(2)));
typedef float v4f __attribute__((ext_vector_type(4)));
typedef float v8f __attribute__((ext_vector_type(8)));

// workspace layout (float offsets)
#define W_IN_OFF   0u        // [32][2048] folded input weight  (65536)
#define B_IN_OFF   65536u    // [32]       folded input bias
#define WO_OFF     65568u    // [2048][32] up_w @ pout_w        (65536)
#define B_O_OFF    131104u   // [2048]     up_w @ pout_b + up_b
#define TABLE_OFF  133152u   // [256][2048] clipped output table (524288)
#define IDX_WS_OFF 657440u   // [16384]    int indices for gather kernel

#define KCHUNK     512       // K tile staged in LDS per step
#define LDS_BPAD   516       // 512 + 4 pad -> row stride = 4 banks (conflict-free)

// ---------------------------------------------------------------------------
// Fold W_in = pin_w @ down_w  [32,2048];  b_in = pin_w @ down_b + pin_b [32]
// ---------------------------------------------------------------------------
__global__ void k_fold_in(const float* __restrict__ down_w, const float* __restrict__ down_b,
                          const float* __restrict__ pin_w, const float* __restrict__ pin_b,
                          float* __restrict__ ws) {
    int t = blockIdx.x * blockDim.x + threadIdx.x;
    if (t < CB_DIM * IN_FEAT) {
        int c = t / IN_FEAT, i = t % IN_FEAT;          // consecutive threads -> consecutive i
        float acc = 0.f;
        for (int h = 0; h < HIDDEN; ++h)
            acc = fmaf(pin_w[c * HIDDEN + h], down_w[h * IN_FEAT + i], acc);
        ws[W_IN_OFF + t] = acc;
    } else if (t < CB_DIM * IN_FEAT + CB_DIM) {
        int c = t - CB_DIM * IN_FEAT;
        float acc = pin_b[c];
        for (int h = 0; h < HIDDEN; ++h)
            acc = fmaf(pin_w[c * HIDDEN + h], down_b[h], acc);
        ws[B_IN_OFF + c] = acc;
    }
}

// ---------------------------------------------------------------------------
// Fold Wo = up_w @ pout_w [2048,32];  b_o = up_w @ pout_b + up_b [2048]
// ---------------------------------------------------------------------------
__global__ void k_fold_out(const float* __restrict__ up_w, const float* __restrict__ up_b,
                           const float* __restrict__ pout_w, const float* __restrict__ pout_b,
                           float* __restrict__ ws) {
    int t = blockIdx.x * blockDim.x + threadIdx.x;
    if (t < OUT_FEAT * CB_DIM) {
        int o = t / CB_DIM, c = t % CB_DIM;
        float acc = 0.f;
        for (int h = 0; h < HIDDEN; ++h)
            acc = fmaf(up_w[o * HIDDEN + h], pout_w[h * CB_DIM + c], acc);
        ws[WO_OFF + t] = acc;
    } else if (t < OUT_FEAT * CB_DIM + OUT_FEAT) {
        int o = t - OUT_FEAT * CB_DIM;
        float acc = up_b[o];
        for (int h = 0; h < HIDDEN; ++h)
            acc = fmaf(up_w[o * HIDDEN + h], pout_b[h], acc);
        ws[B_O_OFF + o] = acc;
    }
}

// ---------------------------------------------------------------------------
// out_table[n][o] = clip(codebook[n] . Wo[o] + b_o[o], -1, 1); commit_loss = 0
// ---------------------------------------------------------------------------
__global__ void k_table(const float* __restrict__ codebook, float* __restrict__ ws,
                        float* __restrict__ commit) {
    int t = blockIdx.x * blockDim.x + threadIdx.x;
    if (t < NUM_CODES * OUT_FEAT) {
        int n = t / OUT_FEAT, o = t % OUT_FEAT;
        const float* wo = ws + WO_OFF + (size_t)o * CB_DIM;
        float acc = ws[B_O_OFF + o];
        #pragma unroll
        for (int c = 0; c < CB_DIM; ++c)
            acc = fmaf(codebook[n * CB_DIM + c], wo[c], acc);
        ws[TABLE_OFF + t] = fminf(1.f, fmaxf(-1.f, acc));
    }
    if (t == 0) commit[0] = 0.0f;
}

// ---------------------------------------------------------------------------
// VQ kernel: per wave, 16 tokens.
//   z[16,32] = x_tile @ W_in^T + b_in  via V_WMMA_F32_16X16X4_F32,
//   B operand fed from LDS (W_in staged in K-chunks, padded conflict-free),
//   then codebook argmin -> indices.
// ---------------------------------------------------------------------------
#define WAVES_PER_BLOCK 8
__global__ __launch_bounds__(256) void k_vq(const float* __restrict__ x,
                                            const float* __restrict__ codebook,
                                            const float* __restrict__ ws,
                                            int* __restrict__ out_idx,
                                            int* __restrict__ ws_idx) {
    __shared__ float wshB[CB_DIM * LDS_BPAD];          // 66 KB staged W_in chunk
    __shared__ float zsh[WAVES_PER_BLOCK][16][32];
    __shared__ float bsc[WAVES_PER_BLOCK][32];
    __shared__ int   bix[WAVES_PER_BLOCK][32];

    const int lane = threadIdx.x & 31;
    const int wave = threadIdx.x >> 5;
    const int tokenBase = (blockIdx.x * WAVES_PER_BLOCK + wave) * 16;

    // A layout: lanes 0-15 rows M (K = k0,k0+1); lanes 16-31 rows M (K = k0+2,k0+3)
    const int mrow = lane & 15;
    const int koff = (lane >> 4) << 1;   // 0 or 2
    const float* xA = x + (size_t)(tokenBase + mrow) * IN_FEAT + koff;

    // B from LDS: per-lane row = lane&15 (+16 for second N-tile), offset k+koff
    const float* bp0 = wshB + (size_t)mrow * LDS_BPAD + koff;
    const float* bp1 = bp0 + (size_t)16 * LDS_BPAD;

    // staging assignment: each thread copies 64 contiguous floats of one W_in row
    const int srow = threadIdx.x >> 3;        // 0..31
    const int sseg = (threadIdx.x & 7) * 64;  // 0,64,...,448

    v8f c0 = {}; v8f c1 = {};
    for (int kc = 0; kc < IN_FEAT; kc += KCHUNK) {
        __syncthreads();   // previous chunk's readers done
        {
            const v4f* src = (const v4f*)(ws + W_IN_OFF + (size_t)srow * IN_FEAT + kc + sseg);
            v4f* dst = (v4f*)(wshB + (size_t)srow * LDS_BPAD + sseg);
            #pragma unroll
            for (int i = 0; i < 16; ++i) dst[i] = src[i];
        }
        __syncthreads();

        const float* xk = xA + kc;
        #pragma unroll 8
        for (int k = 0; k < KCHUNK; k += 4) {
            v2f a  = __builtin_nontemporal_load((const v2f*)(xk + k));  // x read-once: NT
            v2f b0 = *(const v2f*)(bp0 + k);
            v2f b1 = *(const v2f*)(bp1 + k);
            c0 = __builtin_amdgcn_wmma_f32_16x16x4_f32(false, a, false, b0, (short)0, c0, false, false);
            c1 = __builtin_amdgcn_wmma_f32_16x16x4_f32(false, a, false, b1, (short)0, c1, false, false);
        }
    }

    // C/D layout: VGPR r, lanes 0-15 -> M=r, N=lane; lanes 16-31 -> M=r+8, N=lane-16
    const int ncol = lane & 15;
    const int mhi  = (lane >> 4) << 3;   // 0 or 8
    const float bin0 = ws[B_IN_OFF + ncol];
    const float bin1 = ws[B_IN_OFF + ncol + 16];
    #pragma unroll
    for (int r = 0; r < 8; ++r) {
        zsh[wave][r + mhi][ncol]      = c0[r] + bin0;
        zsh[wave][r + mhi][ncol + 16] = c1[r] + bin1;
    }
    __syncthreads();

    // argmin: lane half 0 scans codes 0..127, half 1 scans 128..255
    const int tok  = lane & 15;
    const int half = lane >> 4;
    float zreg[CB_DIM];
    #pragma unroll
    for (int c = 0; c < CB_DIM; ++c) zreg[c] = zsh[wave][tok][c];

    float best = 3.4e38f;
    int   bidx = half * 128;
    const float* cb = codebook + (size_t)half * 128 * CB_DIM;
    for (int n = 0; n < 128; ++n) {
        float s = 0.f;
        #pragma unroll
        for (int c = 0; c < CB_DIM; ++c) {
            float cv = cb[n * CB_DIM + c];
            s = fmaf(cv, cv, s);
            s = fmaf(-2.f * zreg[c], cv, s);
        }
        if (s < best) { best = s; bidx = n + half * 128; }
    }
    bsc[wave][lane] = best;
    bix[wave][lane] = bidx;
    __syncthreads();

    if (lane < 16) {
        float blo = bsc[wave][lane], bhi = bsc[wave][lane + 16];
        int   ilo = bix[wave][lane], ihi = bix[wave][lane + 16];
        int sel = (bhi < blo) ? ihi : ilo;   // tie -> lower index (matches jnp.argmin)
        out_idx[tokenBase + lane] = sel;
        ws_idx[tokenBase + lane]  = sel;
    }
}

// ---------------------------------------------------------------------------
// Streaming gather: out[token] = out_table[idx[token]]  (NT stores, full occ.)
// one block per token row: 2048 floats = 512 float4 across 256 threads
// ---------------------------------------------------------------------------
__global__ __launch_bounds__(256) void k_gather(const float* __restrict__ ws,
                                                const int* __restrict__ ws_idx,
                                                float* __restrict__ out) {
    const int token = blockIdx.x;
    const int idx = ws_idx[token];
    const v4f* src = (const v4f*)(ws + TABLE_OFF + (size_t)idx * OUT_FEAT);
    v4f* dst = (v4f*)(out + (size_t)token * OUT_FEAT);
    #pragma unroll
    for (int j = threadIdx.x; j < OUT_FEAT / 4; j += 256) {
        v4f v = src[j];                         // table stays L2-resident (reused)
        __builtin_nontemporal_store(v, dst + j); // 134 MB stream: bypass cache
    }
}

// ---------------------------------------------------------------------------
extern "C" void kernel_launch(void* const* d_in, const int* in_sizes, int n_in,
                              void* d_out, int out_size, void* d_ws, size_t ws_size,
                              hipStream_t stream) {
    const float* x        = (const float*)d_in[0];
    const float* down_w   = (const float*)d_in[1];
    const float* down_b   = (const float*)d_in[2];
    const float* pin_w    = (const float*)d_in[3];
    const float* pin_b    = (const float*)d_in[4];
    const float* codebook = (const float*)d_in[5];
    const float* pout_w   = (const float*)d_in[6];
    const float* pout_b   = (const float*)d_in[7];
    const float* up_w     = (const float*)d_in[8];
    const float* up_b     = (const float*)d_in[9];

    float* ws  = (float*)d_ws;
    float* out = (float*)d_out;
    int*   out_idx = (int*)((float*)d_out + (size_t)TOKENS * OUT_FEAT);
    float* commit  = (float*)d_out + (size_t)TOKENS * OUT_FEAT + TOKENS;
    int*   ws_idx  = (int*)(ws + IDX_WS_OFF);

    k_fold_in <<<(CB_DIM * IN_FEAT + CB_DIM + 255) / 256, 256, 0, stream>>>(down_w, down_b, pin_w, pin_b, ws);
    k_fold_out<<<(OUT_FEAT * CB_DIM + OUT_FEAT + 255) / 256, 256, 0, stream>>>(up_w, up_b, pout_w, pout_b, ws);
    k_table   <<<(NUM_CODES * OUT_FEAT + 255) / 256, 256, 0, stream>>>(codebook, ws, commit);
    k_vq      <<<TOKENS / (16 * WAVES_PER_BLOCK), 256, 0, stream>>>(x, codebook, ws, out_idx, ws_idx);
    k_gather  <<<TOKENS, 256, 0, stream>>>(ws, ws_idx, out);
}